// GCN_63823214018912
// MI455X (gfx1250) — compile-verified
//
#include <hip/hip_runtime.h>
#include <hip/hip_bf16.h>
#include <cstdint>

typedef float v2f __attribute__((ext_vector_type(2)));
typedef float v8f __attribute__((ext_vector_type(8)));

#define IN_CH  256
#define HID_CH 128
#define KTILE  64
#define PSTRIDE 288   // floats per K-pair row in LDS; 288 % 64 == 32 -> conflict-free b64 reads

__device__ __forceinline__ void atomAddF(float* p, float v) {
    __hip_atomic_fetch_add(p, v, __ATOMIC_RELAXED, __HIP_MEMORY_SCOPE_AGENT);
}

// ---------------- degree / norm ----------------
__global__ void k_init_deg(float* __restrict__ deg, int n) {
    int i = blockIdx.x * blockDim.x + threadIdx.x;
    if (i < n) deg[i] = 1.0f;                     // self-loop
}

__global__ void k_deg_accum(const int* __restrict__ dst, float* __restrict__ deg, int ne) {
    int e = blockIdx.x * blockDim.x + threadIdx.x;
    if (e < ne) atomAddF(&deg[dst[e]], 1.0f);
}

__global__ void k_rsqrt(float* __restrict__ deg, int n) {
    int i = blockIdx.x * blockDim.x + threadIdx.x;
    if (i < n) deg[i] = rsqrtf(deg[i]);           // deg >= 1 always
}

// ---------------- GEMM1: h = x @ W1 via V_WMMA_F32_16X16X4_F32 ----------------
// block = 256 threads = 8 waves; wave w computes rows [blk*128 + w*16, +16) x all 128 cols.
// W1 K-chunk staged in LDS with K-pairs interleaved so each B fragment is one ds_load_b64.
__global__ __launch_bounds__(256) void k_gemm1_wmma(
    const float* __restrict__ x, const float* __restrict__ W1,
    float* __restrict__ h, int n_nodes)
{
    __shared__ float sW[(KTILE / 2) * PSTRIDE];   // 36.8 KB

    const int tid  = threadIdx.x;
    const int wave = tid >> 5;
    const int lane = tid & 31;
    const int half = lane >> 4;                   // 0: K={0,1}, 1: K={2,3}
    const int l16  = lane & 15;

    const int row_base = (blockIdx.x * 8 + wave) * 16;
    int arow = row_base + l16;
    if (arow >= n_nodes) arow = n_nodes - 1;      // clamp reads; keep EXEC full for WMMA
    const float* aptr = x + (size_t)arow * IN_CH + half * 2;

    v8f acc[8] = {};                              // 8 N-tiles of the 16x16 C layout

    for (int k0 = 0; k0 < IN_CH; k0 += KTILE) {
        __syncthreads();
        // stage W1[k0..k0+KTILE)[0..127] pair-interleaved:
        //   sW[p*PSTRIDE + 2*c + j] = W1[k0 + 2p + j][c]
        for (int i = tid; i < (KTILE / 2) * 32; i += 256) {
            int p  = i >> 5;
            int c4 = (i & 31) * 4;
            float4 r0 = *(const float4*)(W1 + (size_t)(k0 + 2 * p + 0) * HID_CH + c4);
            float4 r1 = *(const float4*)(W1 + (size_t)(k0 + 2 * p + 1) * HID_CH + c4);
            float* dp = &sW[p * PSTRIDE + c4 * 2];
            ((float2*)dp)[0] = make_float2(r0.x, r1.x);
            ((float2*)dp)[1] = make_float2(r0.y, r1.y);
            ((float2*)dp)[2] = make_float2(r0.z, r1.z);
            ((float2*)dp)[3] = make_float2(r0.w, r1.w);
        }
        __syncthreads();

#pragma unroll 4
        for (int k = 0; k < KTILE; k += 4) {
            v2f a = *(const v2f*)(aptr + k0 + k);             // (M=l16, K=k+2*half+{0,1})
            const int p = (k >> 1) + half;                    // this half's K-pair row
            const float* bbase = &sW[p * PSTRIDE + l16 * 2];
#pragma unroll
            for (int nt = 0; nt < 8; ++nt) {
                v2f b = *(const v2f*)(bbase + nt * 32);       // (K=pair, N=nt*16+l16), one b64
                acc[nt] = __builtin_amdgcn_wmma_f32_16x16x4_f32(
                    false, a, false, b, (short)0, acc[nt], false, false);
            }
        }
    }

    // C/D layout: VGPR v -> (M = v + 8*half, N = nt*16 + l16)
#pragma unroll
    for (int nt = 0; nt < 8; ++nt) {
#pragma unroll
        for (int v = 0; v < 8; ++v) {
            int r = row_base + v + 8 * half;
            if (r < n_nodes) h[(size_t)r * HID_CH + nt * 16 + l16] = acc[nt][v];
        }
    }
}

// ---------------- layer-1 aggregation ----------------
__global__ void k_agg1_init(const float* __restrict__ h, const float* __restrict__ b1,
                            const float* __restrict__ dinv, float* __restrict__ out1,
                            int n_nodes)
{
    int idx = blockIdx.x * blockDim.x + threadIdx.x;      // one float4 each
    if (idx >= n_nodes * (HID_CH / 4)) return;
    int node = idx >> 5;
    int c = (idx & 31) * 4;
    float di = dinv[node], w = di * di;                   // self-loop norm
    float4 hv = *(const float4*)(h + (size_t)node * HID_CH + c);
    float4 bv = *(const float4*)(b1 + c);
    float4 o; o.x = bv.x + hv.x * w; o.y = bv.y + hv.y * w;
    o.z = bv.z + hv.z * w; o.w = bv.w + hv.w * w;
    *(float4*)(out1 + (size_t)node * HID_CH + c) = o;
}

// one 32-lane group per edge; each lane handles 4 channels
__global__ __launch_bounds__(256) void k_agg1_edges(
    const float* __restrict__ h, const int* __restrict__ src, const int* __restrict__ dst,
    const float* __restrict__ dinv, float* __restrict__ out1, int ne)
{
    int e = blockIdx.x * (blockDim.x >> 5) + (threadIdx.x >> 5);
    if (e >= ne) return;
    int lane = threadIdx.x & 31;
    int s = src[e], d = dst[e];
    float norm = dinv[s] * dinv[d];
    float4 hv = *(const float4*)(h + (size_t)s * HID_CH + lane * 4);
    float* op = out1 + (size_t)d * HID_CH + lane * 4;
    atomAddF(op + 0, hv.x * norm);
    atomAddF(op + 1, hv.y * norm);
    atomAddF(op + 2, hv.z * norm);
    atomAddF(op + 3, hv.w * norm);
}

// ---------------- layer 2: relu -> @W2 -> bias + self-loop into d_out ----------------
__global__ void k_gemm2(const float* __restrict__ out1, const float* __restrict__ W2,
                        const float* __restrict__ b2, const float* __restrict__ dinv,
                        float* __restrict__ h2, float* __restrict__ out, int n_nodes)
{
    int i = blockIdx.x * blockDim.x + threadIdx.x;
    if (i >= n_nodes) return;
    const float* row = out1 + (size_t)i * HID_CH;
    float s0 = 0.f, s1 = 0.f;
#pragma unroll 8
    for (int c = 0; c < HID_CH; c += 4) {
        float4 hv = *(const float4*)(row + c);
        float r0 = fmaxf(hv.x, 0.f), r1 = fmaxf(hv.y, 0.f);
        float r2 = fmaxf(hv.z, 0.f), r3 = fmaxf(hv.w, 0.f);
        s0 += r0 * W2[(c + 0) * 2 + 0] + r1 * W2[(c + 1) * 2 + 0]
            + r2 * W2[(c + 2) * 2 + 0] + r3 * W2[(c + 3) * 2 + 0];
        s1 += r0 * W2[(c + 0) * 2 + 1] + r1 * W2[(c + 1) * 2 + 1]
            + r2 * W2[(c + 2) * 2 + 1] + r3 * W2[(c + 3) * 2 + 1];
    }
    float di = dinv[i], w = di * di;
    h2[i * 2 + 0] = s0;
    h2[i * 2 + 1] = s1;
    out[i * 2 + 0] = b2[0] + s0 * w;   // fully initializes d_out before edge scatter
    out[i * 2 + 1] = b2[1] + s1 * w;
}

__global__ void k_agg2_edges(const float* __restrict__ h2, const int* __restrict__ src,
                             const int* __restrict__ dst, const float* __restrict__ dinv,
                             float* __restrict__ out, int ne)
{
    int e = blockIdx.x * blockDim.x + threadIdx.x;
    if (e >= ne) return;
    int s = src[e], d = dst[e];
    float norm = dinv[s] * dinv[d];
    atomAddF(&out[d * 2 + 0], h2[s * 2 + 0] * norm);
    atomAddF(&out[d * 2 + 1], h2[s * 2 + 1] * norm);
}

// ---------------- launch ----------------
extern "C" void kernel_launch(void* const* d_in, const int* in_sizes, int n_in,
                              void* d_out, int out_size, void* d_ws, size_t ws_size,
                              hipStream_t stream)
{
    const float* x  = (const float*)d_in[0];
    const int*   ei = (const int*)d_in[1];
    const float* W1 = (const float*)d_in[2];
    const float* b1 = (const float*)d_in[3];
    const float* W2 = (const float*)d_in[4];
    const float* b2 = (const float*)d_in[5];
    float* out = (float*)d_out;

    const int n_nodes = in_sizes[0] / IN_CH;
    const int n_edges = in_sizes[1] / 2;
    const int* src = ei;
    const int* dst = ei + n_edges;

    // workspace layout (4 KB aligned): dinv | h (aliased by h2) | out1
    char* ws = (char*)d_ws;
    size_t off = 0;
    float* dinv = (float*)(ws + off);
    off += ((size_t)n_nodes * 4 + 4095) & ~(size_t)4095;
    float* h = (float*)(ws + off);
    size_t hb = ((size_t)n_nodes * HID_CH * 4 + 4095) & ~(size_t)4095;
    off += hb;
    float* out1 = (float*)(ws + off);
    float* h2 = h;   // h is dead once k_agg1_edges completes

    k_init_deg <<<(n_nodes + 255) / 256, 256, 0, stream>>>(dinv, n_nodes);
    k_deg_accum<<<(n_edges + 255) / 256, 256, 0, stream>>>(dst, dinv, n_edges);
    k_rsqrt    <<<(n_nodes + 255) / 256, 256, 0, stream>>>(dinv, n_nodes);

    k_gemm1_wmma<<<(n_nodes + 127) / 128, 256, 0, stream>>>(x, W1, h, n_nodes);

    int t1 = n_nodes * (HID_CH / 4);
    k_agg1_init <<<(t1 + 255) / 256, 256, 0, stream>>>(h, b1, dinv, out1, n_nodes);
    k_agg1_edges<<<(n_edges + 7) / 8, 256, 0, stream>>>(h, src, dst, dinv, out1, n_edges);

    k_gemm2     <<<(n_nodes + 255) / 256, 256, 0, stream>>>(out1, W2, b2, dinv, h2, out, n_nodes);
    k_agg2_edges<<<(n_edges + 255) / 256, 256, 0, stream>>>(h2, src, dst, dinv, out, n_edges);
}